// C_Loss_52338471469527
// MI455X (gfx1250) — compile-verified
//
#include <hip/hip_runtime.h>
#include <math.h>

typedef __attribute__((ext_vector_type(2))) float v2f;
typedef __attribute__((ext_vector_type(8))) float v8f;

#define EPSF 1e-6f

// Exact 32-lane sum using V_WMMA_F32_16X16X4_F32 with B = ones.
// A layout (32-bit A 16x4): lane L<16: VGPR0=(M=L,K=0), VGPR1=(M=L,K=1);
//                           lane L>=16: VGPR0=(M=L-16,K=2), VGPR1=(M=L-16,K=3).
// With a.y=0: D[m][n] = v_m + v_{m+16} for every n (ones-B is layout-invariant).
// D layout: VGPR r holds row r (lanes 0-15) and row r+8 (lanes 16-31), so a
// lane-local sum of the 8 D regs gives rows 0-7 (lanes<16) / rows 8-15
// (lanes>=16); one shfl_xor(16) completes the full sum on every lane.
__device__ __forceinline__ float wave_sum_wmma(float v) {
    v2f a; a.x = v;    a.y = 0.0f;
    v2f b; b.x = 1.0f; b.y = 1.0f;
    v8f c = {};
    v8f d = __builtin_amdgcn_wmma_f32_16x16x4_f32(
        /*neg_a=*/false, a, /*neg_b=*/false, b,
        /*c_mod=*/(short)0, c, /*reuse_a=*/false, /*reuse_b=*/false);
    float s = d[0] + d[1] + d[2] + d[3] + d[4] + d[5] + d[6] + d[7];
    s += __shfl_xor(s, 16, 32);
    return s;
}

// acc layout: [0]=bce_sum, [1]=fd_num, [2]=fd_cnt, [3..66]=dsum64, [67..130]=cnt64
__global__ void zero_acc(float* acc) {
    int i = threadIdx.x;
    if (i < 131) acc[i] = 0.0f;
}

__global__ __launch_bounds__(256) void loss_main(const float* __restrict__ pred,
                                                 const float* __restrict__ gt,
                                                 float* __restrict__ acc) {
    const int b   = blockIdx.x;   // 0..4095 : n*64 + patch
    const int n   = b >> 6;
    const int pid = b & 63;       // 64x64 patch id within image (8x8 grid)
    const int py  = pid >> 3, px = pid & 7;
    const int t   = threadIdx.x;  // 256 threads: one 4x4 pooled cell each
    const int cy  = t >> 4, cx = t & 15;
    const int h0  = py * 64 + cy * 4;
    const int w0  = px * 64 + cx * 4;
    const size_t base = (size_t)n * 262144u + (size_t)h0 * 512u + (size_t)w0;

    float bce = 0.0f, dsum = 0.0f, gsum = 0.0f, ssum = 0.0f;
#pragma unroll
    for (int r = 0; r < 4; ++r) {
        const float4 p4 = *(const float4*)(pred + base + (size_t)r * 512u);
        const float4 g4 = *(const float4*)(gt   + base + (size_t)r * 512u);
        const float pv[4] = {p4.x, p4.y, p4.z, p4.w};
        const float gv[4] = {g4.x, g4.y, g4.z, g4.w};
#pragma unroll
        for (int i = 0; i < 4; ++i) {
            const float p = pv[i];
            const float g = gv[i];
            // gt is exactly 0/1: gt*log(p)+(1-gt)*log(1-p) == log(gt ? p : 1-p)
            bce += __logf(g > 0.5f ? p : 1.0f - p);
            const float s = __builtin_amdgcn_rcpf(1.0f + __expf(-p)); // sigmoid
            dsum += fabsf(s - g);
            gsum += g;
            ssum += s;
        }
    }
    // FD loss per pooled 4x4 cell: S0 = mean(sig), S1 = mean(gt).
    // The (S>0)&(S<16) mask is a no-op here (means in [0,1]; S1*(S1>0)==S1).
    const float s0  = ssum * 0.0625f;
    const float s1  = gsum * 0.0625f;
    float fdn = fabsf(s0 - s1);
    float fdc = (gsum > 0.0f) ? 1.0f : 0.0f;

    // Wave-level exact sums via WMMA (EXEC all-ones: no divergence yet).
    bce  = wave_sum_wmma(bce);
    dsum = wave_sum_wmma(dsum);
    gsum = wave_sum_wmma(gsum);
    fdn  = wave_sum_wmma(fdn);
    fdc  = wave_sum_wmma(fdc);

    __shared__ float red[8][5];
    const int wave = t >> 5, lane = t & 31;
    if (lane == 0) {
        red[wave][0] = bce;
        red[wave][1] = dsum;
        red[wave][2] = gsum;
        red[wave][3] = fdn;
        red[wave][4] = fdc;
    }
    __syncthreads();
    if (t < 5) {
        float v = 0.0f;
#pragma unroll
        for (int w = 0; w < 8; ++w) v += red[w][t];
        if      (t == 0) atomicAdd(&acc[0], v);
        else if (t == 1) atomicAdd(&acc[3 + pid], v);
        else if (t == 2) atomicAdd(&acc[67 + pid], v);
        else if (t == 3) atomicAdd(&acc[1], v);
        else             atomicAdd(&acc[2], v);
    }
}

__global__ void finish(const float* __restrict__ acc, float* __restrict__ out) {
    if (threadIdx.x != 0 || blockIdx.x != 0) return;
    const float* d64 = acc + 3;
    const float* c64 = acc + 67;

    // ps = 64 : 8x8 patches, ps*N = 4096
    float t64 = 0.0f;
    for (int i = 0; i < 64; ++i) {
        const float c = d64[i] / fmaxf(c64[i], EPSF);
        t64 += 4096.0f * c * c;
    }
    // ps = 128 : 4x4 patches (2x2 aggregation), ps*N = 8192
    float d128[16], c128[16];
    for (int I = 0; I < 4; ++I)
        for (int J = 0; J < 4; ++J) {
            float D = 0.0f, C = 0.0f;
            for (int a = 0; a < 2; ++a)
                for (int bb = 0; bb < 2; ++bb) {
                    const int idx = (2 * I + a) * 8 + (2 * J + bb);
                    D += d64[idx]; C += c64[idx];
                }
            d128[I * 4 + J] = D; c128[I * 4 + J] = C;
        }
    float t128 = 0.0f;
    for (int i = 0; i < 16; ++i) {
        const float c = d128[i] / fmaxf(c128[i], EPSF);
        t128 += 8192.0f * c * c;
    }
    // ps = 256 : 2x2 patches, ps*N = 16384
    float t256 = 0.0f;
    for (int I = 0; I < 2; ++I)
        for (int J = 0; J < 2; ++J) {
            float D = 0.0f, C = 0.0f;
            for (int a = 0; a < 2; ++a)
                for (int bb = 0; bb < 2; ++bb) {
                    const int idx = (2 * I + a) * 4 + (2 * J + bb);
                    D += d128[idx]; C += c128[idx];
                }
            const float c = D / fmaxf(C, EPSF);
            t256 += 16384.0f * c * c;
        }

    // size_t = sqrt(nh*nw*ps*ps) = 512 for all levels; N = 64
    const float l64  = fminf(fmaxf(sqrtf(t64)  * (1.0f / (512.0f * 64.0f)), 0.0f), 1.0f);
    const float l128 = fminf(fmaxf(sqrtf(t128) * (1.0f / (512.0f * 64.0f)), 0.0f), 1.0f);
    const float l256 = fminf(fmaxf(sqrtf(t256) * (1.0f / (512.0f * 64.0f)), 0.0f), 1.0f);
    const float loss_patch = (l64 + l128 + l256) / 64.0f;  // / num_patches_last

    // FD: c = fd_num / max(fd_cnt, eps); vessel = sqrt(N * 512 * c^2)
    const float cfd     = acc[1] / fmaxf(acc[2], EPSF);
    const float vessel  = sqrtf(64.0f * 512.0f * cfd * cfd);
    const float size_tt = sqrtf(349440.0f);  // sum of (2^k)^2, k=9..4
    const float loss_fd = vessel / size_tt / 64.0f;

    const float bce = -acc[0] / 16777216.0f;  // mean over 64*512*512

    out[0] = 1.1f * bce + 0.02f * loss_fd + 0.03f * loss_patch;
}

extern "C" void kernel_launch(void* const* d_in, const int* in_sizes, int n_in,
                              void* d_out, int out_size, void* d_ws, size_t ws_size,
                              hipStream_t stream) {
    const float* pred = (const float*)d_in[0];
    const float* gt   = (const float*)d_in[1];
    float* acc = (float*)d_ws;   // 131 floats of scratch accumulators
    float* out = (float*)d_out;

    zero_acc<<<1, 256, 0, stream>>>(acc);
    loss_main<<<4096, 256, 0, stream>>>(pred, gt, acc);
    finish<<<1, 32, 0, stream>>>(acc, out);
}